// MultiTableEmbeddings_57260503990934
// MI455X (gfx1250) — compile-verified
//
#include <hip/hip_runtime.h>
#include <stdint.h>

// Multi-table embedding gather for MI455X (gfx1250).
// out[b, t, :] = tables[t, idx[b,t], :],  B=16384, T=26, D=128, fp32.
// Pure bandwidth problem (~440 MB -> ~19us floor at 23.3 TB/s).
//
// Hot kernel is branch-free: each wave32 moves 8 embedding rows (8 x 512B)
// through the CDNA5 async global<->LDS datapath (ASYNCcnt). Data never
// touches VGPRs; indices are fetched with scalar loads (immediate offsets
// off one uniform base); stores are non-temporal so the write-once output
// does not evict table rows from L2. Ragged batch edge (never hit for
// B=16384) is handled by a separate guarded tail kernel.

#define NUM_TABLES      26
#define EMB_D           128
#define ROW_BYTES       (EMB_D * 4)            // 512B per embedding row
#define ROWS_PER_WAVE   8
#define WAVES_PER_BLOCK 8
#define ROWS_PER_BLOCK  (ROWS_PER_WAVE * WAVES_PER_BLOCK)   // 64 batch rows/block

typedef __attribute__((address_space(3))) float lds_float;

__global__ __launch_bounds__(256)
void emb_gather_main(const int* __restrict__ cat,      // [B, T]
                     const float* __restrict__ tables, // [T, V, D]
                     float* __restrict__ out,          // [B, T, D]
                     long vocab)
{
    // 8 waves x 8 rows x 512B = 32 KB LDS per block.
    __shared__ float smem[WAVES_PER_BLOCK * ROWS_PER_WAVE * EMB_D];

    // readfirstlane makes the wave id provably uniform -> scalar index loads.
    const int wave = __builtin_amdgcn_readfirstlane((int)(threadIdx.x >> 5));
    const int lane = (int)(threadIdx.x & 31);

    const int t  = blockIdx.y;
    const int b0 = (blockIdx.x * WAVES_PER_BLOCK + wave) * ROWS_PER_WAVE;

    // LDS byte offset of this lane's 16B slot in this wave's first row slot.
    lds_float* lbase = (lds_float*)smem;
    const unsigned lds0 =
        (unsigned)(uintptr_t)(&lbase[(wave * ROWS_PER_WAVE) * EMB_D + lane * 4]);

    // Uniform base + small immediate offsets -> clause of s_load_b32.
    const int* __restrict__ cp = cat + ((size_t)b0 * NUM_TABLES + t);
    long idx[ROWS_PER_WAVE];
#pragma unroll
    for (int r = 0; r < ROWS_PER_WAVE; ++r)
        idx[r] = (long)cp[r * NUM_TABLES];

    // 64-bit bases; only the lane*16B part is per-lane.
    const unsigned long long table_base = (unsigned long long)(uintptr_t)
        (tables + (long)t * vocab * EMB_D + (long)lane * 4);
    const unsigned long long out_base = (unsigned long long)(uintptr_t)
        (out + ((long)b0 * NUM_TABLES + t) * EMB_D + (long)lane * 4);

    // Phase 1: 8 async gathers, memory -> LDS (ASYNCcnt), no VGPR data.
#pragma unroll
    for (int r = 0; r < ROWS_PER_WAVE; ++r) {
        const unsigned long long g_src =
            table_base + (unsigned long long)(idx[r]) * (unsigned long long)ROW_BYTES;
        const unsigned ldsaddr = lds0 + (unsigned)(r * ROW_BYTES);
        asm volatile("global_load_async_to_lds_b128 %0, %1, off"
                     :
                     : "v"(ldsaddr), "v"(g_src)
                     : "memory");
    }

    // All 8 LDS row slots populated.
    asm volatile("s_wait_asynccnt 0" ::: "memory");

    // Phase 2: 8 async stores, LDS -> memory (non-temporal, write-once data).
#pragma unroll
    for (int r = 0; r < ROWS_PER_WAVE; ++r) {
        const unsigned long long g_dst =
            out_base + (unsigned long long)r * (unsigned long long)(NUM_TABLES * ROW_BYTES);
        const unsigned ldsaddr = lds0 + (unsigned)(r * ROW_BYTES);
        asm volatile("global_store_async_from_lds_b128 %0, %1, off th:TH_STORE_NT"
                     :
                     : "v"(g_dst), "v"(ldsaddr)
                     : "memory");
    }
    // s_endpgm performs an implicit wait-idle, covering outstanding stores.
}

// Guarded tail: one row per wave, launched only when batch % 64 != 0.
__global__ __launch_bounds__(256)
void emb_gather_tail(const int* __restrict__ cat,
                     const float* __restrict__ tables,
                     float* __restrict__ out,
                     long vocab, int b_start, int batch)
{
    __shared__ float smem[WAVES_PER_BLOCK * EMB_D];

    const int wave = __builtin_amdgcn_readfirstlane((int)(threadIdx.x >> 5));
    const int lane = (int)(threadIdx.x & 31);

    const int t = blockIdx.y;
    const int b = b_start + blockIdx.x * WAVES_PER_BLOCK + wave;
    if (b >= batch) return;   // uniform per wave

    const long idx = (long)cat[(size_t)b * NUM_TABLES + t];

    const float* src = tables + ((long)t * vocab + idx) * EMB_D + (long)lane * 4;
    float*       dst = out    + ((long)b * NUM_TABLES + t) * EMB_D + (long)lane * 4;

    lds_float* lbase = (lds_float*)smem;
    const unsigned ldsaddr =
        (unsigned)(uintptr_t)(&lbase[wave * EMB_D + lane * 4]);

    const unsigned long long g_src = (unsigned long long)(uintptr_t)src;
    const unsigned long long g_dst = (unsigned long long)(uintptr_t)dst;

    asm volatile("global_load_async_to_lds_b128 %0, %1, off"
                 :: "v"(ldsaddr), "v"(g_src) : "memory");
    asm volatile("s_wait_asynccnt 0" ::: "memory");
    asm volatile("global_store_async_from_lds_b128 %0, %1, off th:TH_STORE_NT"
                 :: "v"(g_dst), "v"(ldsaddr) : "memory");
}

extern "C" void kernel_launch(void* const* d_in, const int* in_sizes, int n_in,
                              void* d_out, int out_size, void* d_ws, size_t ws_size,
                              hipStream_t stream) {
    const int*   cat    = (const int*)d_in[0];    // categorical_inputs [B, T]
    const float* tables = (const float*)d_in[1];  // tables [T, V, D]
    float*       out    = (float*)d_out;          // [B, T, D]

    const int  batch = in_sizes[0] / NUM_TABLES;                       // 16384
    const long vocab = (long)in_sizes[1] / ((long)NUM_TABLES * EMB_D); // 50000

    const int batch_main = batch & ~(ROWS_PER_BLOCK - 1);  // multiple of 64

    if (batch_main > 0) {
        dim3 grid(batch_main / ROWS_PER_BLOCK, NUM_TABLES);
        emb_gather_main<<<grid, 256, 0, stream>>>(cat, tables, out, vocab);
    }
    const int tail = batch - batch_main;
    if (tail > 0) {
        dim3 grid((tail + WAVES_PER_BLOCK - 1) / WAVES_PER_BLOCK, NUM_TABLES);
        emb_gather_tail<<<grid, 256, 0, stream>>>(cat, tables, out, vocab,
                                                  batch_main, batch);
    }
}